// TrainableButterfly_25855703121979
// MI455X (gfx1250) — compile-verified
//
#include <hip/hip_runtime.h>
#include <math.h>

#define LENGTH 2048
#define NLEV 11
#define BATCH 4096
#define TPB 256
#define PADN 2334  // padidx(2047) = 2333

typedef float v2f __attribute__((ext_vector_type(2)));
typedef float v4f __attribute__((ext_vector_type(4)));

// LDS padding: e -> e + e/8 + e/64 (conflict-free bit-reversal scatter).
__device__ __forceinline__ int padidx(int e) { return e + (e >> 3) + (e >> 6); }

// CDNA5 async copy engine: per-lane global source (GVS: sgpr64 base + vgpr32
// offset), per-lane LDS destination byte address in the vdst VGPR.
__device__ __forceinline__ void async_ld_b32(unsigned lds_byte, unsigned voff,
                                             const void* sbase) {
  asm volatile("global_load_async_to_lds_b32 %0, %1, %2"
               :
               : "v"(lds_byte), "v"(voff), "s"(sbase)
               : "memory");
}

__device__ __forceinline__ void wait_async0() {
#if __has_builtin(__builtin_amdgcn_s_wait_asynccnt)
  __builtin_amdgcn_s_wait_asynccnt(0);
#else
  asm volatile("s_wait_asynccnt 0" ::: "memory");
#endif
}

__device__ __forceinline__ float2 cmul(float2 a, float2 b) {
  return make_float2(a.x * b.x - a.y * b.y, a.x * b.y + a.y * b.x);
}

// Packed complex multiply: a * w (w = complex constant), vector form so the
// backend can select v_pk_mul_f32/v_pk_fma_f32 with opsel/neg modifiers.
__device__ __forceinline__ v2f cmulv(v2f a, v2f w) {
  v2f wj = {-w.y, w.x}; // i-rotated w (folds to opsel+neg_lo)
  return w * a.x + wj * a.y;
}

// Fused butterfly with K = T: u = (T wa)*a, v = (iT wb)*b.
// out0 = u + v ; out1 = i*(u - v) = (v.y-u.y, u.x-v.x).
__device__ __forceinline__ void bfly(v2f a, v2f b, v2f wA, v2f wB,
                                     v2f& o0, v2f& o1) {
  v2f u = cmulv(a, wA);
  v2f v = cmulv(b, wB);
  o0 = u + v;
  o1 = (v2f){v.y - u.y, u.x - v.x};
}

// Inter-level permutation of level l: element m -> g + 2p + q.
__device__ __forceinline__ int newpos(int m, int l) {
  int bs = 4 << l, half = 2 << l;
  return (m & ~(bs - 1)) + ((m & (half - 1)) << 1) + ((m & (bs - 1)) >> (l + 1));
}

// ---------------------------------------------------------------------------
// Table layout in workspace (v4f units), 12288 * 16 B = 196,608 bytes:
//   l in 0..9 : Wt[l*1024 + n] = (T*wa | i*T*wb)
//   l == 10   : Wt[10240 + 2n] = (C00|C01) ; Wt[10240 + 2n+1] = (C10|C11)
// wa/wb = exp(i*phases[l][2n,2n+1]) * (l>=1 ? CROSS[l-1][2n,2n+1] : 1)
// C00=T*(p*wa), C01=iT*(p*wb), C10=iT*(q*wa), C11=T*(q*wb),
// p,q = exp(i*phases[11][2n,2n+1]).
// ---------------------------------------------------------------------------

// Main: one block per batch row.  Async bit-reversal scatter into LDS, then
// 5 radix-4 passes (levels 2p, 2p+1 in registers, exploiting j(m+1)=j(m)+2 and
// j(2(n+2^l))=j(2n)+1), then level 10 fused with the final twiddle.
__global__ __launch_bounds__(TPB) void bf_main(const float* __restrict__ xr,
                                               const float* __restrict__ xi,
                                               const v4f* __restrict__ Wt,
                                               float2* __restrict__ out) {
  __shared__ v2f smem[2 * PADN];
  v2f* bufA = smem;
  v2f* bufB = smem + PADN;

  const int b = blockIdx.x;
  const int t = threadIdx.x;
  const float* rowr = xr + (size_t)b * LENGTH;
  const float* rowi = xi + (size_t)b * LENGTH;

  // Bit-reversal scatter: element e (coalesced read) lands at brev11(e).
#pragma unroll
  for (int k = 0; k < 8; ++k) {
    int e = t + k * TPB;
    int rev = (int)(__brev((unsigned)e) >> 21);
    unsigned lbyte = (unsigned)(size_t)(&bufA[padidx(rev)]);
    async_ld_b32(lbyte, (unsigned)(e * 4), rowr);
    async_ld_b32(lbyte + 4, (unsigned)(e * 4), rowi);
  }
  wait_async0();
  __syncthreads();

  // 5 passes, each doing levels (2p, 2p+1) in registers.
#pragma unroll
  for (int p = 0; p < 5; ++p) {
    const int l = 2 * p;
    const v2f* src = (p & 1) ? bufB : bufA;
    v2f* dst = (p & 1) ? bufA : bufB;
#pragma unroll
    for (int k = 0; k < 2; ++k) {
      int q = t + k * TPB;                                  // quad id, 0..511
      int n = ((q >> l) << (l + 1)) | (q & ((1 << l) - 1)); // pair, bit l == 0
      int n2 = n + (1 << l);                                // partner pair
      v2f a0 = src[padidx(2 * n)];
      v2f b0 = src[padidx(2 * n + 1)];
      v2f a1 = src[padidx(2 * n2)];
      v2f b1 = src[padidx(2 * n2 + 1)];

      // level l
      v4f w0 = Wt[(l << 10) + n];
      v4f w1 = Wt[(l << 10) + n2];
      v2f o0, o1, o0p, o1p;
      bfly(a0, b0, w0.xy, w0.zw, o0, o1);
      bfly(a1, b1, w1.xy, w1.zw, o0p, o1p);

      // outputs occupy new elements j0..j0+3 = level-(l+1) pairs P0, P0+1
      int j0 = newpos(2 * n, l); // even
      int P0 = j0 >> 1;

      // level l+1 (in registers)
      v4f wz0 = Wt[((l + 1) << 10) + P0];
      v4f wz1 = Wt[((l + 1) << 10) + P0 + 1];
      v2f z00, z01, z10, z11;
      bfly(o0, o0p, wz0.xy, wz0.zw, z00, z01);
      bfly(o1, o1p, wz1.xy, wz1.zw, z10, z11);

      int J0 = newpos(j0, l + 1);
      int J1 = newpos(j0 + 2, l + 1);
      dst[padidx(J0)] = z00;
      dst[padidx(J0 + 2)] = z01;
      dst[padidx(J1)] = z10;
      dst[padidx(J1 + 2)] = z11;
    }
    __syncthreads();
  }

  // Level 10 (data in bufB after pass 4), final twiddle folded, b128 stores.
  v4f* orow = (v4f*)(out + (size_t)b * LENGTH);
#pragma unroll
  for (int k = 0; k < 4; ++k) {
    int n = t + k * TPB;
    v2f a = bufB[padidx(2 * n)];
    v2f bv = bufB[padidx(2 * n + 1)];
    v4f c0 = Wt[10240 + 2 * n];
    v4f c1 = Wt[10240 + 2 * n + 1];
    v2f acc0 = cmulv(a, c0.xy) + cmulv(bv, c0.zw);
    v2f acc1 = cmulv(a, c1.xy) + cmulv(bv, c1.zw);
    orow[n] = (v4f){acc0.x, acc0.y, acc1.x, acc1.y};
  }
}

// Prep: build the fused constant tables (one thread per (level, pair)).
__global__ __launch_bounds__(TPB) void bf_prep(const float* __restrict__ phases,
                                               float4* __restrict__ Wt) {
  int tid = blockIdx.x * blockDim.x + threadIdx.x; // 0 .. 11*1024-1
  int l = tid >> 10;
  int n = tid & 1023;
  int j0 = 2 * n, j1 = j0 + 1;

  float pa = phases[l * LENGTH + j0];
  float pb = phases[l * LENGTH + j1];
  float sa, ca, sb, cb;
  __sincosf(pa, &sa, &ca);
  __sincosf(pb, &sb, &cb);
  float2 wa = make_float2(ca, sa);
  float2 wb = make_float2(cb, sb);

  if (l >= 1) {
    int lev = l - 1;
    int half = 2 << lev;
    int bs = half << 1;
    int n0 = (j0 & (bs - 1)) >> 1;              // q=0 -> p
    int n1 = half - 1 - ((j1 & (bs - 1)) >> 1); // q=1 -> half-1-p
    float f0 = (float)n0, f1 = (float)n1;
    float a0 = __expf(f0 * -0.01005033585350145f); // ln(0.99)
    float a1 = __expf(f1 * -0.01005033585350145f);
    float s0, c0, s1, c1;
    __sincosf(0.01f * f0, &s0, &c0);
    __sincosf(0.01f * f1, &s1, &c1);
    wa = cmul(wa, make_float2(a0 * c0, a0 * s0));
    wb = cmul(wb, make_float2(a1 * c1, a1 * s1));
  }

  const float Tc = 0.70710678118654752f;
  if (l < 10) {
    Wt[(l << 10) + n] = make_float4(Tc * wa.x, Tc * wa.y, -Tc * wb.y, Tc * wb.x);
  } else {
    float p11a = phases[11 * LENGTH + j0];
    float p11b = phases[11 * LENGTH + j1];
    float sp, cp, sq, cq;
    __sincosf(p11a, &sp, &cp);
    __sincosf(p11b, &sq, &cq);
    float2 P = make_float2(cp, sp), Q = make_float2(cq, sq);
    float2 pa2 = cmul(P, wa), pb2 = cmul(P, wb);
    float2 qa2 = cmul(Q, wa), qb2 = cmul(Q, wb);
    Wt[10240 + j0] = make_float4(Tc * pa2.x, Tc * pa2.y, -Tc * pb2.y, Tc * pb2.x);
    Wt[10240 + j1] = make_float4(-Tc * qa2.y, Tc * qa2.x, Tc * qb2.x, Tc * qb2.y);
  }
}

extern "C" void kernel_launch(void* const* d_in, const int* in_sizes, int n_in,
                              void* d_out, int out_size, void* d_ws, size_t ws_size,
                              hipStream_t stream) {
  const float* xr = (const float*)d_in[0];
  const float* xi = (const float*)d_in[1];
  const float* ph = (const float*)d_in[2];
  float4* Wt = (float4*)d_ws; // 12288 * 16 B = 196,608 bytes

  bf_prep<<<dim3((11 * 1024) / TPB), dim3(TPB), 0, stream>>>(ph, Wt);
  bf_main<<<dim3(BATCH), dim3(TPB), 0, stream>>>(xr, xi, (const v4f*)Wt,
                                                 (float2*)d_out);
}